// ESN_36550171689161
// MI455X (gfx1250) — compile-verified
//
#include <hip/hip_runtime.h>
#include <hip/hip_bf16.h>
#include <math.h>

// ---------------------------------------------------------------------------
// ESN: ifft( (scan(tanh, fft(X) @ W_in^T, diag d)) @ W_out^T ) / T
// T=262144 (2^18), IN=32, R=50, OUT=32.  complex64 throughout after the FFT.
// W_in / W_out are real -> each complex GEMM = 2 real GEMMs sharing the B
// fragment -> V_WMMA_F32_16X16X4_F32 with dual accumulators.
// All boundary handling is branch-free (index clamp + mask-multiply) so the
// WMMA waves never touch EXEC.
// ---------------------------------------------------------------------------

#define ESN_T    262144
#define ESN_IN   32
#define ESN_R    50
#define ESN_OUT  32
#define FFT_LOG2 18
#define PI_F     3.14159265358979323846f

typedef float v2f __attribute__((ext_vector_type(2)));
typedef float v8f __attribute__((ext_vector_type(8)));

// ---------------------------------------------------------------------------
// real input -> interleaved complex
// ---------------------------------------------------------------------------
__global__ void esn_real_to_complex(const float* __restrict__ X,
                                    float2* __restrict__ out, int n) {
    int i = blockIdx.x * blockDim.x + threadIdx.x;
    if (i < n) out[i] = make_float2(X[i], 0.0f);
}

// ---------------------------------------------------------------------------
// One radix-2 Stockham pass over all 32 columns.
// Layout: (t, c) -> idx t*32 + c (float2).  dir = -1 forward, +1 inverse.
// ---------------------------------------------------------------------------
__global__ void esn_fft_pass(const float2* __restrict__ src,
                             float2* __restrict__ dst,
                             int l, int m, float dir) {
    int tid = blockIdx.x * blockDim.x + threadIdx.x;
    int c = tid & (ESN_IN - 1);
    int b = tid >> 5;                 // butterfly id in [0, l*m)
    if (b >= l * m) return;
    int k = b % m;
    int j = b / m;

    float ang = dir * (2.0f * PI_F) * (float)j / (float)(2 * l);
    float wi, wr;
    __sincosf(ang, &wi, &wr);

    float2 x0 = src[(size_t)(k + j * m) * ESN_IN + c];
    float2 x1 = src[(size_t)(k + j * m + l * m) * ESN_IN + c];

    float2 sum = make_float2(x0.x + x1.x, x0.y + x1.y);
    float2 dif = make_float2(x0.x - x1.x, x0.y - x1.y);
    float2 tw  = make_float2(wr * dif.x - wi * dif.y,
                             wr * dif.y + wi * dif.x);

    dst[(size_t)(k + 2 * j * m) * ESN_IN + c]     = sum;
    dst[(size_t)(k + 2 * j * m + m) * ESN_IN + c] = tw;
}

// ---------------------------------------------------------------------------
// proj[t,r] = sum_k Xc[t,k] * W_in[r,k]   (complex A, real shared B)
// One wave per 16x16 tile; K=32 in 8 chunks of 4.
// A frag (16x4 f32): lane L holds A[L&15][v + 2*(L>>4)], v=0..1
// B frag (4x16 f32): lane L holds B[v + 2*(L>>4)][L&15]
// C/D (16x16 f32):   lane L, vgpr i -> C[i + 8*(L>>4)][L&15]
// ---------------------------------------------------------------------------
__global__ void esn_gemm_proj(const float2* __restrict__ Xc,
                              const float* __restrict__ Win,
                              float2* __restrict__ proj) {
    const int t0   = blockIdx.x * 16;
    const int r0   = blockIdx.y * 16;
    const int lane = threadIdx.x;
    const int half = lane >> 4;
    const int row  = lane & 15;
    const int rcol = r0 + row;

    // B fragments, hoisted & branch-free: clamp row, mask-multiply to zero.
    const float  bmask = (rcol < ESN_R) ? 1.0f : 0.0f;
    const int    rclmp = (rcol < ESN_R) ? rcol : (ESN_R - 1);
    const float* wrow  = Win + (size_t)rclmp * ESN_IN;
    v2f bw[8];
#pragma unroll
    for (int c = 0; c < 8; ++c) {
        const float2 w = *(const float2*)(wrow + 4 * c + 2 * half);
        bw[c][0] = w.x * bmask;
        bw[c][1] = w.y * bmask;
    }

    const float2* arow = Xc + (size_t)(t0 + row) * ESN_IN;

    v8f accR = {};
    v8f accI = {};
#pragma unroll
    for (int c = 0; c < 8; ++c) {
        // two adjacent complex values -> one b128 load
        const float4 x = *(const float4*)(arow + 4 * c + 2 * half);
        v2f aR = { x.x, x.z };
        v2f aI = { x.y, x.w };
        accR = __builtin_amdgcn_wmma_f32_16x16x4_f32(
            false, aR, false, bw[c], (short)0, accR, false, false);
        accI = __builtin_amdgcn_wmma_f32_16x16x4_f32(
            false, aI, false, bw[c], (short)0, accI, false, false);
    }

    if (rcol < ESN_R) {
#pragma unroll
        for (int i = 0; i < 8; ++i) {
            int t = t0 + i + 8 * half;
            proj[(size_t)t * ESN_R + rcol] = make_float2(accR[i], accI[i]);
        }
    }
}

// ---------------------------------------------------------------------------
// Diagonal reservoir scan: 50 independent complex recurrences.
//   z = p_t + d*s ;  s = ctanh(z) = (tanh x + i tan y)/(1 + i tanh x tan y)
// ---------------------------------------------------------------------------
__global__ void esn_scan(const float2* __restrict__ proj,
                         const float* __restrict__ dres,
                         float2* __restrict__ S) {
    const int r = threadIdx.x;
    if (r >= ESN_R) return;
    const float d = dres[r];
    float sr = 0.0f, si = 0.0f;
    for (int t = 0; t < ESN_T; ++t) {
        __builtin_prefetch(&proj[(size_t)(t + 96) * ESN_R + r], 0, 1);
        float2 p = proj[(size_t)t * ESN_R + r];
        float zr = fmaf(d, sr, p.x);
        float zi = fmaf(d, si, p.y);
        float tx = tanhf(zr);
        float ty = tanf(zi);
        float tx2 = tx * tx;
        float ty2 = ty * ty;
        float inv = 1.0f / (1.0f + tx2 * ty2);
        sr = tx * (1.0f + ty2) * inv;
        si = ty * (1.0f - tx2) * inv;
        S[(size_t)t * ESN_R + r] = make_float2(sr, si);
    }
}

// ---------------------------------------------------------------------------
// out[t,o] = sum_k S[t,k] * W_out[o,k]  (complex A, real B), K=50 -> 13 chunks
// Tail chunk handled by clamping the load base to 48 (stays inside row t)
// and zeroing masked elements -- no EXEC manipulation anywhere.
// ---------------------------------------------------------------------------
__global__ void esn_gemm_out(const float2* __restrict__ S,
                             const float* __restrict__ Wout,
                             float2* __restrict__ Yc) {
    const int t0   = blockIdx.x * 16;
    const int o0   = blockIdx.y * 16;
    const int lane = threadIdx.x;
    const int half = lane >> 4;
    const int row  = lane & 15;
    const int ocol = o0 + row;

    const float* worow = Wout + (size_t)ocol * ESN_R;

    v2f bw[13];
#pragma unroll
    for (int c = 0; c < 13; ++c) {
        const int   k  = 4 * c + 2 * half;
        const int   kc = (k > ESN_R - 2) ? (ESN_R - 2) : k;   // clamp to 48
        const float m0 = (k     < ESN_R) ? 1.0f : 0.0f;
        const float m1 = (k + 1 < ESN_R) ? 1.0f : 0.0f;
        const float2 w = *(const float2*)(worow + kc);
        bw[c][0] = w.x * m0;
        bw[c][1] = w.y * m1;
    }

    const float2* srow = S + (size_t)(t0 + row) * ESN_R;

    v8f accR = {};
    v8f accI = {};
#pragma unroll
    for (int c = 0; c < 13; ++c) {
        const int   k  = 4 * c + 2 * half;
        const int   kc = (k > ESN_R - 2) ? (ESN_R - 2) : k;
        const float m0 = (k     < ESN_R) ? 1.0f : 0.0f;
        const float m1 = (k + 1 < ESN_R) ? 1.0f : 0.0f;
        const float4 s = *(const float4*)(srow + kc);         // 2 complex
        v2f aR = { s.x * m0, s.z * m1 };
        v2f aI = { s.y * m0, s.w * m1 };
        accR = __builtin_amdgcn_wmma_f32_16x16x4_f32(
            false, aR, false, bw[c], (short)0, accR, false, false);
        accI = __builtin_amdgcn_wmma_f32_16x16x4_f32(
            false, aI, false, bw[c], (short)0, accI, false, false);
    }

#pragma unroll
    for (int i = 0; i < 8; ++i) {
        int t = t0 + i + 8 * half;
        Yc[(size_t)t * ESN_OUT + ocol] = make_float2(accR[i], accI[i]);
    }
}

// ---------------------------------------------------------------------------
// final 1/T scale into d_out
// ---------------------------------------------------------------------------
__global__ void esn_scale_out(const float2* __restrict__ src,
                              float2* __restrict__ dst, float s, int n) {
    int i = blockIdx.x * blockDim.x + threadIdx.x;
    if (i < n) {
        float2 v = src[i];
        dst[i] = make_float2(v.x * s, v.y * s);
    }
}

// ---------------------------------------------------------------------------
extern "C" void kernel_launch(void* const* d_in, const int* in_sizes, int n_in,
                              void* d_out, int out_size, void* d_ws, size_t ws_size,
                              hipStream_t stream) {
    const float* X    = (const float*)d_in[0];   // [T, IN]
    const float* Win  = (const float*)d_in[1];   // [R, IN]
    const float* dres = (const float*)d_in[2];   // [R]
    const float* Wout = (const float*)d_in[3];   // [OUT, R]

    // workspace layout (float2):
    //  bufA [T*32] | bufB [T*32] | proj [T*50] | states [T*50]   (~328 MB)
    float2* bufA   = (float2*)d_ws;
    float2* bufB   = bufA + (size_t)ESN_T * ESN_IN;
    float2* proj   = bufB + (size_t)ESN_T * ESN_IN;
    float2* states = proj + (size_t)ESN_T * ESN_R;

    const int nTC       = ESN_T * ESN_IN;           // T*32 complex elements
    const int cvtBlocks = (nTC + 255) / 256;
    const int bfThreads = (ESN_T / 2) * ESN_IN;     // butterflies * columns
    const int bfBlocks  = (bfThreads + 255) / 256;

    // 1) real -> complex
    esn_real_to_complex<<<cvtBlocks, 256, 0, stream>>>(X, bufA, nTC);

    // 2) forward FFT: 18 Stockham radix-2 passes, ping-pong (ends in bufA)
    {
        int l = ESN_T / 2, m = 1;
        float2 *a = bufA, *b = bufB;
        for (int p = 0; p < FFT_LOG2; ++p) {
            esn_fft_pass<<<bfBlocks, 256, 0, stream>>>(a, b, l, m, -1.0f);
            float2* t = a; a = b; b = t;
            l >>= 1; m <<= 1;
        }
    }

    // 3) proj = Xc @ W_in^T   (WMMA, one wave per 16x16 tile)
    esn_gemm_proj<<<dim3(ESN_T / 16, 4), 32, 0, stream>>>(bufA, Win, proj);

    // 4) diagonal reservoir scan (50 lanes, serial over T)
    esn_scan<<<1, 64, 0, stream>>>(proj, dres, states);

    // 5) out = states @ W_out^T  (WMMA) -> bufA as [T,32] complex
    esn_gemm_out<<<dim3(ESN_T / 16, 2), 32, 0, stream>>>(states, Wout, bufA);

    // 6) inverse FFT: 18 passes, ping-pong (ends in bufA)
    {
        int l = ESN_T / 2, m = 1;
        float2 *a = bufA, *b = bufB;
        for (int p = 0; p < FFT_LOG2; ++p) {
            esn_fft_pass<<<bfBlocks, 256, 0, stream>>>(a, b, l, m, +1.0f);
            float2* t = a; a = b; b = t;
            l >>= 1; m <<= 1;
        }
    }

    // 7) 1/T scale into d_out (interleaved complex64)
    esn_scale_out<<<cvtBlocks, 256, 0, stream>>>(bufA, (float2*)d_out,
                                                 1.0f / (float)ESN_T, nTC);
}